// AlignmentLoss_8065948582478
// MI455X (gfx1250) — compile-verified
//
#include <hip/hip_runtime.h>
#include <hip/hip_bf16.h>
#include <cfloat>

// Problem constants from the reference: B=16, C=1024, T=16, H=14, W=14
#define B_ 16
#define C_ 1024
#define N_ 3136            // T*H*W = 16*14*14
#define TILES_PER_BATCH 196  // N_/16
#define WAVES_PER_BLOCK 4
#define BLOCKS_PER_BATCH 49  // 196/4

typedef __attribute__((ext_vector_type(2))) float v2f;
typedef __attribute__((ext_vector_type(8))) float v8f;

// ---------------------------------------------------------------------------
// Kernel 1: batched matvec via V_WMMA_F32_16X16X4_F32.
// Per wave: one (batch b, 16-wide n-tile). D = A(16x4) * B(4x16) accumulated
// over K = C = 1024 (256 WMMA steps).
//   A row0 = pos[b, k..k+3], row1 = neg[b, k..k+3], rows 2..15 = 0
//   B[k][n] = ft[b, k, n0+n]
//   D vgpr0 lanes0-15 = posdot[n0..n0+15], vgpr1 lanes0-15 = negdot.
// f32 A 16x4 layout: lane(0-15)=M, lanes16-31 repeat M; VGPR0/1 hold
// K = {0,1} (low half) / {2,3} (high half).
// f32 B 4x16 layout: lane&15 = N; low half K={0,1}, high half K={2,3}.
// ---------------------------------------------------------------------------
__global__ __launch_bounds__(128) void dots_wmma_kernel(
    const float* __restrict__ ft, const float* __restrict__ pos,
    const float* __restrict__ neg, float* __restrict__ wsPos,
    float* __restrict__ wsNeg) {
  __shared__ __align__(16) float ldsA[2 * C_];  // row0: pos[b,:], row1: neg[b,:]

  const int bb = blockIdx.x / BLOCKS_PER_BATCH;
  const int blkInB = blockIdx.x % BLOCKS_PER_BATCH;

  // Stage this batch's pos/neg rows into LDS (8 KB) once per block.
  for (int i = threadIdx.x; i < C_; i += 128) {
    ldsA[i] = pos[bb * C_ + i];
    ldsA[C_ + i] = neg[bb * C_ + i];
  }
  __syncthreads();

  const int wave = threadIdx.x >> 5;
  const int lane = threadIdx.x & 31;
  const int tile = blkInB * WAVES_PER_BLOCK + wave;
  const int n0 = tile * 16;

  const int m = lane & 15;          // M row for A, N column for B
  const int khalf = lane >> 4;      // 0 -> K{0,1}, 1 -> K{2,3}
  const int mr = (m < 2) ? m : 0;   // clamp so every lane's LDS address is valid
  const bool amask = (m < 2);       // only rows 0 (pos) and 1 (neg) are non-zero

  const float* fmb = ft + (size_t)bb * C_ * N_;
  // Per-lane base for B fragment: row (2*khalf), column n0+m.
  const float* bbase = fmb + (size_t)(2 * khalf) * N_ + n0 + m;
  const float* abase = &ldsA[mr * C_ + 2 * khalf];

  v8f acc = {};  // zero accumulator

#pragma unroll 4
  for (int k = 0; k < C_; k += 4) {
    // A fragment: two consecutive f32 from LDS (8B aligned: even index).
    float2 av = *reinterpret_cast<const float2*>(abase + k);
    v2f a;
    a.x = amask ? av.x : 0.0f;
    a.y = amask ? av.y : 0.0f;

    // B fragment: rows k+2*khalf and k+2*khalf+1, column n0+m.
    const float* p = bbase + (size_t)k * N_;
    v2f b;
    b.x = p[0];
    b.y = p[N_];

    // D = A*B + acc  (8 args: neg_a, A, neg_b, B, c_mod, C, reuse_a, reuse_b)
    acc = __builtin_amdgcn_wmma_f32_16x16x4_f32(
        false, a, false, b, (short)0, acc, false, false);
  }

  // D layout: VGPR r, lanes 0-15 => M=r, N=lane. Row 0 = posdot, row 1 = negdot.
  if (lane < 16) {
    wsPos[(size_t)bb * N_ + n0 + lane] = acc[0];
    wsNeg[(size_t)bb * N_ + n0 + lane] = acc[1];
  }
}

// ---------------------------------------------------------------------------
// Kernel 2: per-batch softmax(posdot*1000) + margin-loss partial.
// One block per batch; classic LDS tree reductions.
// ---------------------------------------------------------------------------
__global__ __launch_bounds__(256) void softmax_loss_kernel(
    const float* __restrict__ wsPos, const float* __restrict__ wsNeg,
    float* __restrict__ attn, float* __restrict__ wsLoss) {
  const int b = blockIdx.x;
  const int tid = threadIdx.x;
  __shared__ float redA[256];
  __shared__ float redB[256];

  const float* pd = wsPos + (size_t)b * N_;
  const float* nd = wsNeg + (size_t)b * N_;

  // Pass 1: max over N
  float lmax = -FLT_MAX;
  for (int i = tid; i < N_; i += 256) lmax = fmaxf(lmax, pd[i]);
  redA[tid] = lmax;
  __syncthreads();
  for (int s = 128; s > 0; s >>= 1) {
    if (tid < s) redA[tid] = fmaxf(redA[tid], redA[tid + s]);
    __syncthreads();
  }
  const float smax = redA[0];
  __syncthreads();

  // Pass 2: Z = sum exp, S = sum lossmap*exp
  float zsum = 0.0f, lsum = 0.0f;
  for (int i = tid; i < N_; i += 256) {
    float p = pd[i];
    float e = __expf((p - smax) * 1000.0f);  // /TEMPERATURE
    zsum += e;
    float lm = fmaxf(0.2f - 100000.0f * (p - nd[i]), 0.0f);  // MARGIN, SCALE
    lsum += lm * e;
  }
  redA[tid] = zsum;
  redB[tid] = lsum;
  __syncthreads();
  for (int s = 128; s > 0; s >>= 1) {
    if (tid < s) {
      redA[tid] += redA[tid + s];
      redB[tid] += redB[tid + s];
    }
    __syncthreads();
  }
  const float invZ = 1.0f / redA[0];

  // Pass 3: write attention map
  for (int i = tid; i < N_; i += 256)
    attn[(size_t)b * N_ + i] = __expf((pd[i] - smax) * 1000.0f) * invZ;

  if (tid == 0) wsLoss[b] = redB[0] * invZ;
}

// ---------------------------------------------------------------------------
// Kernel 3: reduce 16 per-batch loss partials, /B.
// ---------------------------------------------------------------------------
__global__ void reduce_loss_kernel(const float* __restrict__ wsLoss,
                                   float* __restrict__ loss) {
  if (threadIdx.x == 0) {
    float s = 0.0f;
    for (int b = 0; b < B_; ++b) s += wsLoss[b];
    *loss = s * (1.0f / (float)B_);
  }
}

extern "C" void kernel_launch(void* const* d_in, const int* in_sizes, int n_in,
                              void* d_out, int out_size, void* d_ws,
                              size_t ws_size, hipStream_t stream) {
  const float* ft = (const float*)d_in[0];   // (16,1024,16,14,14) f32
  const float* pos = (const float*)d_in[1];  // (16,1024) f32
  const float* neg = (const float*)d_in[2];  // (16,1024) f32

  float* out = (float*)d_out;
  float* attn = out;              // 16*3136 = 50176 floats
  float* loss = out + B_ * N_;    // 1 float

  float* wsPos = (float*)d_ws;           // B*N floats
  float* wsNeg = wsPos + B_ * N_;        // B*N floats
  float* wsLoss = wsNeg + B_ * N_;       // B floats
  (void)in_sizes; (void)n_in; (void)out_size; (void)ws_size;

  dots_wmma_kernel<<<B_ * BLOCKS_PER_BATCH, 128, 0, stream>>>(
      ft, pos, neg, wsPos, wsNeg);
  softmax_loss_kernel<<<B_, 256, 0, stream>>>(wsPos, wsNeg, attn, wsLoss);
  reduce_loss_kernel<<<1, 32, 0, stream>>>(wsLoss, loss);
}